// GaussianBlur_44796508897414
// MI455X (gfx1250) — compile-verified
//
#include <hip/hip_runtime.h>
#include <hip/hip_bf16.h>

typedef float v2f __attribute__((ext_vector_type(2)));
typedef float v8f __attribute__((ext_vector_type(8)));

#define KS    21
#define PAD   10
#define IMG   512
#define TM    32         // output rows per workgroup
#define TN    64         // output cols per workgroup (4 waves x 16)
#define LROWS 64         // 4 full 16-row WMMA chunks (covers TM + 2*PAD = 52)
#define LCOLS 84         // TN + 2*PAD
#define LSTR  84         // LDS row stride in floats (conflict-free patterns)
#define NSTEP 9          // K = 36 = 9 * 4

__device__ __forceinline__ int refl(int i, int n) {
    // jnp 'reflect' (mirror, edge not repeated); PAD < n so one fold suffices
    i = (i < 0) ? -i : i;
    i = (i >= n) ? (2 * n - 2 - i) : i;
    return i;
}

__global__ __launch_bounds__(128) void GaussianBlur_44796508897414_kernel(
        const float* __restrict__ x, float* __restrict__ y) {
    __shared__ float s_in[LROWS * LSTR];   // raw input tile (rows i0-10 .. i0+53)
    __shared__ float s_h [LROWS * LSTR];   // horizontally filtered tile
    __shared__ float s_w [64];             // zero-padded 1D weights: s_w[16+t] = w[t]

    const int tid  = threadIdx.x;
    const int lane = tid & 31;
    const int wave = tid >> 5;
    const int half = lane >> 4;   // 0: lanes 0-15, 1: lanes 16-31
    const int l16  = lane & 15;

    const int j0  = blockIdx.x * TN;   // output col base of workgroup
    const int i0  = blockIdx.y * TM;   // output row base
    const int img = blockIdx.z;        // flattened (batch*channel)

    const float* __restrict__ xin  = x + (size_t)img * IMG * IMG;
    float* __restrict__       yout = y + (size_t)img * IMG * IMG;

    // ---- zero-padded 1D gaussian weight table in LDS ----
    // w1d[t] = exp(-(t-10)^2/(2*25)) / sum  (separable factor of the 2D kernel)
    if (tid < 64) {
        float s = 0.f;
        #pragma unroll
        for (int t = 0; t < KS; ++t) {
            float d = (float)t - 10.f;
            s += __expf(-d * d * 0.02f);
        }
        const int t = tid - 16;        // table covers Toeplitz index range [-16, 48)
        float v = 0.f;
        if (t >= 0 && t < KS) {
            float d = (float)t - 10.f;
            v = __expf(-d * d * 0.02f) / s;
        }
        s_w[tid] = v;
    }

    // ---- cooperative load of input tile with reflection ----
    for (int idx = tid; idx < LROWS * LCOLS; idx += 128) {
        const int r  = idx / LCOLS;
        const int c  = idx - r * LCOLS;
        const int gr = refl(i0 - PAD + r, IMG);
        const int gc = refl(j0 - PAD + c, IMG);
        s_in[r * LSTR + c] = xin[gr * IMG + gc];
    }
    __syncthreads();

    // ---- banded-Toeplitz WMMA fragment: value(k, j) = w[k - j], 0 <= k-j < 21 ----
    // f32 A(16x4) and B(4x16) lane layouts are symmetric (m <-> n), so the same
    // register values serve as B in the horizontal pass and A in the vertical pass.
    v2f tw[NSTEP];
    {
        const int bw = 16 + 2 * half - l16;   // in [1, 18]
        #pragma unroll
        for (int st = 0; st < NSTEP; ++st) {
            tw[st].x = s_w[bw + 4 * st + 0];
            tw[st].y = s_w[bw + 4 * st + 1];
        }
    }

    const int c0 = wave * 16;   // this wave's 16-col strip

    // ---- pass 1: horizontal conv via WMMA, 4 row-chunks of 16 ----
    // H[m][c0+n] = sum_k s_in[m][c0+k] * w[k-n]
    #pragma unroll
    for (int rc = 0; rc < 4; ++rc) {
        v8f acc = {0.f, 0.f, 0.f, 0.f, 0.f, 0.f, 0.f, 0.f};
        const float* ap = &s_in[(rc * 16 + l16) * LSTR + c0 + 2 * half];
        #pragma unroll
        for (int st = 0; st < NSTEP; ++st) {
            v2f a = *(const v2f*)(ap + 4 * st);   // 8B-aligned ds_load_b64
            acc = __builtin_amdgcn_wmma_f32_16x16x4_f32(
                      false, a, false, tw[st], (short)0, acc, false, false);
        }
        // D layout: VGPR v, lane L -> M = v + 8*(L/16), N = L%16
        #pragma unroll
        for (int v = 0; v < 8; ++v)
            s_h[(rc * 16 + v + 8 * half) * LSTR + c0 + l16] = acc[v];
    }
    __syncthreads();

    // ---- pass 2: vertical conv via WMMA, 2 interleaved output row-tiles ----
    // out[i0+16*ot+i][j0+c0+n] = sum_k w[k-i] * s_h[16*ot+k][c0+n]
    v8f acc0 = {0.f, 0.f, 0.f, 0.f, 0.f, 0.f, 0.f, 0.f};
    v8f acc1 = {0.f, 0.f, 0.f, 0.f, 0.f, 0.f, 0.f, 0.f};
    #pragma unroll
    for (int st = 0; st < NSTEP; ++st) {
        const int k = 4 * st + 2 * half;
        const float* bp = &s_h[k * LSTR + c0 + l16];
        v2f b0, b1;
        b0.x = bp[0];
        b0.y = bp[LSTR];
        b1.x = bp[16 * LSTR];
        b1.y = bp[17 * LSTR];
        acc0 = __builtin_amdgcn_wmma_f32_16x16x4_f32(
                   false, tw[st], false, b0, (short)0, acc0, false, false);
        acc1 = __builtin_amdgcn_wmma_f32_16x16x4_f32(
                   false, tw[st], false, b1, (short)0, acc1, false, false);
    }
    #pragma unroll
    for (int v = 0; v < 8; ++v) {
        yout[(size_t)(i0 + v + 8 * half) * IMG + j0 + c0 + l16]      = acc0[v];
        yout[(size_t)(i0 + 16 + v + 8 * half) * IMG + j0 + c0 + l16] = acc1[v];
    }
}

extern "C" void kernel_launch(void* const* d_in, const int* in_sizes, int n_in,
                              void* d_out, int out_size, void* d_ws, size_t ws_size,
                              hipStream_t stream) {
    const float* x = (const float*)d_in[0];
    float* y = (float*)d_out;
    const int imgs = in_sizes[0] / (IMG * IMG);       // 32 * 3 = 96
    dim3 grid(IMG / TN, IMG / TM, imgs);              // (8, 16, 96)
    dim3 block(128, 1, 1);                            // 4 wave32
    GaussianBlur_44796508897414_kernel<<<grid, block, 0, stream>>>(x, y);
}